// POIGraph3_38465727103680
// MI455X (gfx1250) — compile-verified
//
#include <hip/hip_runtime.h>
#include <math.h>

#define N_NODES 50000
#define N_EDGES 800000
#define HIDDEN  128
#define H3      384
#define TOPK    15

typedef __attribute__((ext_vector_type(2))) float v2f;
typedef __attribute__((ext_vector_type(8))) float v8f;

// ---------------------------------------------------------------- utilities
__global__ void k_zero_int(int* p, int n) {
    int i = blockIdx.x * blockDim.x + threadIdx.x;
    if (i < n) p[i] = 0;
}

__global__ void k_gather_h0(const int* __restrict__ inputs,
                            const float* __restrict__ emb,
                            float* __restrict__ x) {
    int i = blockIdx.x * blockDim.x + threadIdx.x;
    if (i >= N_NODES * HIDDEN) return;
    int n = i >> 7, j = i & 127;
    x[i] = emb[(size_t)inputs[n] * HIDDEN + j];
}

// ------------------------------------------------------------- degree / CSR
__global__ void k_count(const int* __restrict__ dst, int* __restrict__ counts) {
    int e = blockIdx.x * blockDim.x + threadIdx.x;
    if (e < N_EDGES) atomicAdd(&counts[dst[e]], 1);
}

__global__ void k_dis(const int* __restrict__ counts, float* __restrict__ dis) {
    int n = blockIdx.x * blockDim.x + threadIdx.x;
    if (n < N_NODES) dis[n] = rsqrtf((float)counts[n] + 1.0f);
}

// single-block exclusive scan (N small: 50000)
__global__ void k_scan(const int* __restrict__ counts, int* __restrict__ offsets, int n) {
    __shared__ int sh[1024];
    __shared__ int carry;
    if (threadIdx.x == 0) carry = 0;
    __syncthreads();
    for (int base = 0; base < n; base += 1024) {
        int i = base + threadIdx.x;
        int v = (i < n) ? counts[i] : 0;
        sh[threadIdx.x] = v;
        __syncthreads();
        for (int off = 1; off < 1024; off <<= 1) {
            int t = (threadIdx.x >= off) ? sh[threadIdx.x - off] : 0;
            __syncthreads();
            sh[threadIdx.x] += t;
            __syncthreads();
        }
        if (i < n) offsets[i] = carry + sh[threadIdx.x] - v;   // exclusive
        __syncthreads();
        if (threadIdx.x == 1023) carry += sh[1023];
        __syncthreads();
    }
    if (threadIdx.x == 0) offsets[n] = carry;
}

__global__ void k_fill(const int* __restrict__ dst, const int* __restrict__ offsets,
                       int* __restrict__ fill, int* __restrict__ csr) {
    int e = blockIdx.x * blockDim.x + threadIdx.x;
    if (e >= N_EDGES) return;
    int d = dst[e];
    int p = atomicAdd(&fill[d], 1);
    csr[offsets[d] + p] = e;
}

// deterministic order inside each segment: sort edge ids ascending
__global__ void k_sort_csr(const int* __restrict__ offsets, int* __restrict__ csr) {
    int n = blockIdx.x * blockDim.x + threadIdx.x;
    if (n >= N_NODES) return;
    int beg = offsets[n], end = offsets[n + 1];
    for (int i = beg + 1; i < end; ++i) {
        int key = csr[i];
        int j = i - 1;
        while (j >= beg && csr[j] > key) { csr[j + 1] = csr[j]; --j; }
        csr[j + 1] = key;
    }
}

__global__ void k_enorm(const int* __restrict__ src, const int* __restrict__ dst,
                        const float* __restrict__ dis, float* __restrict__ enorm) {
    int e = blockIdx.x * blockDim.x + threadIdx.x;
    if (e < N_EDGES) enorm[e] = dis[src[e]] * dis[dst[e]];
}

// ----------------------------------------------------------- WMMA fp32 GEMM
// D[M x J] = A[M x 128] * B + bias
// TB: B(k,j) = W[j*128+k] (i.e. A@W^T) ; else B(k,j) = W[k*J+j] (A@W)
// One wave computes a 16-row x 128-col strip. Software-pipelined: the A
// fragment and 8 B fragments of stage k+4 are loaded while the 8 WMMAs of
// stage k execute (2x unrolled so the two register stages alternate).
template <bool TB>
__global__ void k_gemm_wmma(const float* __restrict__ A, const float* __restrict__ B,
                            const float* __restrict__ bias, float* __restrict__ D, int J) {
    const int K = HIDDEN;
    const int lane = threadIdx.x;
    const int half = lane >> 4;
    const int col  = lane & 15;
    const int m16   = blockIdx.x * 16;      // row tile base
    const int ncol0 = blockIdx.y * 128;     // 8 column tiles of 16
    const float* arow = A + (size_t)(m16 + col) * K;

    auto load_a = [&](int k0) -> v2f {
        v2f a; int ka = k0 + 2 * half;
        a.x = arow[ka]; a.y = arow[ka + 1];
        return a;
    };
    auto load_b = [&](int k0, int t) -> v2f {
        v2f b; int ka = k0 + 2 * half;
        int jc = ncol0 + t * 16 + col;
        if (TB) {
            const float* brow = B + (size_t)jc * K;
            b.x = brow[ka]; b.y = brow[ka + 1];
        } else {
            b.x = B[(size_t)ka * J + jc];
            b.y = B[(size_t)(ka + 1) * J + jc];
        }
        return b;
    };

    v8f acc[8];
#pragma unroll
    for (int t = 0; t < 8; ++t) acc[t] = (v8f){};

    // prologue: stage 0
    v2f a0 = load_a(0), a1;
    v2f b0[8], b1[8];
#pragma unroll
    for (int t = 0; t < 8; ++t) b0[t] = load_b(0, t);

#pragma unroll
    for (int k0 = 0; k0 < K; k0 += 8) {
        // issue loads for stage k0+4 while computing stage k0
        a1 = load_a(k0 + 4);
#pragma unroll
        for (int t = 0; t < 8; ++t) b1[t] = load_b(k0 + 4, t);
#pragma unroll
        for (int t = 0; t < 8; ++t)
            acc[t] = __builtin_amdgcn_wmma_f32_16x16x4_f32(
                false, a0, false, b0[t], (short)0, acc[t], false, false);
        // issue loads for stage k0+8 while computing stage k0+4
        if (k0 + 8 < K) {
            a0 = load_a(k0 + 8);
#pragma unroll
            for (int t = 0; t < 8; ++t) b0[t] = load_b(k0 + 8, t);
        }
#pragma unroll
        for (int t = 0; t < 8; ++t)
            acc[t] = __builtin_amdgcn_wmma_f32_16x16x4_f32(
                false, a1, false, b1[t], (short)0, acc[t], false, false);
    }

#pragma unroll
    for (int t = 0; t < 8; ++t) {
        int jc = ncol0 + t * 16 + col;
        float bv = bias ? bias[jc] : 0.0f;
#pragma unroll
        for (int r = 0; r < 8; ++r) {
            int orow = m16 + r + 8 * half;
            D[(size_t)orow * J + jc] = acc[t][r] + bv;
        }
    }
}

// --------------------------------------------------- deterministic pull-agg
// out[n,:] = sum_{e in csr[n]} wt[e] * feat[src[e],:]
__global__ void k_agg(const int* __restrict__ offsets, const int* __restrict__ csr,
                      const int* __restrict__ src, const float* __restrict__ wt,
                      const float* __restrict__ feat, float* __restrict__ out) {
    int node = blockIdx.x * 8 + (threadIdx.x >> 5);
    int lane = threadIdx.x & 31;
    if (node >= N_NODES) return;
    int beg = offsets[node], end = offsets[node + 1];
    float4 acc = make_float4(0.f, 0.f, 0.f, 0.f);
    for (int i = beg; i < end; ++i) {
        int e = csr[i];
        if (i + 1 < end) {  // cover the csr->src->row pointer-chase latency
            int en = csr[i + 1];
            __builtin_prefetch(feat + (size_t)src[en] * HIDDEN + lane * 4, 0, 0);
        }
        float w = wt[e];
        const float4 v = *reinterpret_cast<const float4*>(
            feat + (size_t)src[e] * HIDDEN + lane * 4);
        acc.x += w * v.x; acc.y += w * v.y; acc.z += w * v.z; acc.w += w * v.w;
    }
    *reinterpret_cast<float4*>(out + (size_t)node * HIDDEN + lane * 4) = acc;
}

__global__ void k_gcn_combine(const float* __restrict__ agg, const float* __restrict__ xw,
                              const float* __restrict__ dis, const float* __restrict__ b,
                              float* __restrict__ h) {
    int i = blockIdx.x * blockDim.x + threadIdx.x;
    if (i >= N_NODES * HIDDEN) return;
    int n = i >> 7, j = i & 127;
    float d = dis[n];
    h[i] = fmaxf(agg[i] + d * d * xw[i] + b[j], 0.0f);
}

// ------------------------------------------------------- norms & edge score
__global__ void k_nrm(const float* __restrict__ h, float* __restrict__ nrm) {
    int node = blockIdx.x * 8 + (threadIdx.x >> 5);
    int lane = threadIdx.x & 31;
    if (node >= N_NODES) return;
    const float4 v = *reinterpret_cast<const float4*>(h + (size_t)node * HIDDEN + lane * 4);
    float s = v.x * v.x + v.y * v.y + v.z * v.z + v.w * v.w;
    for (int o = 16; o > 0; o >>= 1) s += __shfl_xor(s, o, 32);
    if (lane == 0) nrm[node] = sqrtf(s);
}

__global__ void k_scores(const int* __restrict__ src, const int* __restrict__ dst,
                         const float* __restrict__ h, const float* __restrict__ nrm,
                         float* __restrict__ scores) {
    int e = blockIdx.x * 8 + (threadIdx.x >> 5);
    int lane = threadIdx.x & 31;
    if (e >= N_EDGES) return;
    int s = src[e], d = dst[e];
    const float4 a = *reinterpret_cast<const float4*>(h + (size_t)s * HIDDEN + lane * 4);
    const float4 b = *reinterpret_cast<const float4*>(h + (size_t)d * HIDDEN + lane * 4);
    float acc = a.x * b.x + a.y * b.y + a.z * b.z + a.w * b.w;
    for (int o = 16; o > 0; o >>= 1) acc += __shfl_xor(acc, o, 32);
    if (lane == 0) {
        const float eps = 1e-8f;
        scores[e] = acc / (fmaxf(nrm[s], eps) * fmaxf(nrm[d], eps));
    }
}

// rank-count top-k (matches lexsort((-score, dst)) + rank < k; ties by edge id)
__global__ void k_topk(const int* __restrict__ offsets, const int* __restrict__ csr,
                       const float* __restrict__ scores, float* __restrict__ keepf) {
    int n = blockIdx.x * blockDim.x + threadIdx.x;
    if (n >= N_NODES) return;
    int beg = offsets[n], end = offsets[n + 1];
    for (int i = beg; i < end; ++i) {
        int e = csr[i];
        float se = scores[e];
        int rank = 0;
        for (int j = beg; j < end; ++j) {
            int e2 = csr[j];
            float s2 = scores[e2];
            rank += (s2 > se) || (s2 == se && e2 < e);
        }
        keepf[e] = (rank < TOPK) ? 1.0f : 0.0f;
    }
}

// ------------------------------------------------------------------ GRU
__global__ void k_gru(const float* __restrict__ gi, const float* __restrict__ gh,
                      const float* __restrict__ xin, float* __restrict__ xout) {
    int i = blockIdx.x * blockDim.x + threadIdx.x;
    if (i >= N_NODES * HIDDEN) return;
    int n = i >> 7, j = i & 127;
    const float* gir = gi + (size_t)n * H3;
    const float* ghr = gh + (size_t)n * H3;
    float r = 1.0f / (1.0f + expf(-(gir[j] + ghr[j])));
    float z = 1.0f / (1.0f + expf(-(gir[HIDDEN + j] + ghr[HIDDEN + j])));
    float nn = tanhf(gir[2 * HIDDEN + j] + r * ghr[2 * HIDDEN + j]);
    xout[i] = (1.0f - z) * nn + z * xin[i];
}

// ------------------------------------------------------------------ driver
extern "C" void kernel_launch(void* const* d_in, const int* in_sizes, int n_in,
                              void* d_out, int out_size, void* d_ws, size_t ws_size,
                              hipStream_t stream) {
    const int*   inputs = (const int*)d_in[0];
    const int*   Aedges = (const int*)d_in[1];
    const float* emb    = (const float*)d_in[2];
    const float* W1     = (const float*)d_in[3];
    const float* b1     = (const float*)d_in[4];
    const float* W2     = (const float*)d_in[5];
    const float* b2     = (const float*)d_in[6];
    const float* Wg     = (const float*)d_in[7];   // [2,128,128]
    const float* Wih    = (const float*)d_in[8];   // [384,128]
    const float* Whh    = (const float*)d_in[9];   // [384,128]
    const float* bih    = (const float*)d_in[10];
    const float* bhh    = (const float*)d_in[11];
    const int* src = Aedges;
    const int* dst = Aedges + N_EDGES;

    const size_t NH  = (size_t)N_NODES * HIDDEN;
    const size_t NH3 = (size_t)N_NODES * H3;
    float* x      = (float*)d_ws;
    float* tmp    = x + NH;
    float* agg    = tmp + NH;
    float* h      = agg + NH;
    float* gi     = h + NH;
    float* gh     = gi + NH3;
    float* dis    = gh + NH3;
    float* nrm    = dis + N_NODES;
    float* scores = nrm + N_NODES;
    float* enorm  = scores + N_EDGES;
    float* keepf  = enorm + N_EDGES;
    int* counts   = (int*)(keepf + N_EDGES);
    int* offsets  = counts + N_NODES;
    int* fill     = offsets + N_NODES + 1;
    int* csr      = fill + N_NODES;

    const int TB256 = 256;
    const int gNH   = (int)((NH + TB256 - 1) / TB256);
    const int gN    = (N_NODES + TB256 - 1) / TB256;
    const int gE    = (N_EDGES + TB256 - 1) / TB256;
    const int gNode8 = (N_NODES + 7) / 8;
    const int gEdge8 = (N_EDGES + 7) / 8;
    const dim3 gemm128(N_NODES / 16, HIDDEN / 128);  // (3125, 1)
    const dim3 gemm384(N_NODES / 16, H3 / 128);      // (3125, 3)

    // --- setup: h0 gather, degrees, CSR
    k_zero_int<<<gN, TB256, 0, stream>>>(counts, N_NODES);
    k_zero_int<<<gN, TB256, 0, stream>>>(fill, N_NODES);
    k_gather_h0<<<gNH, TB256, 0, stream>>>(inputs, emb, x);
    k_count<<<gE, TB256, 0, stream>>>(dst, counts);
    k_dis<<<gN, TB256, 0, stream>>>(counts, dis);
    k_scan<<<1, 1024, 0, stream>>>(counts, offsets, N_NODES);
    k_fill<<<gE, TB256, 0, stream>>>(dst, offsets, fill, csr);
    k_sort_csr<<<gN, TB256, 0, stream>>>(offsets, csr);
    k_enorm<<<gE, TB256, 0, stream>>>(src, dst, dis, enorm);

    // --- GCN layer 1: xw = x@W1^T ; agg = seg_sum(norm*xw[src]) ; relu combine
    k_gemm_wmma<true><<<gemm128, 32, 0, stream>>>(x, W1, nullptr, tmp, HIDDEN);
    k_agg<<<gNode8, TB256, 0, stream>>>(offsets, csr, src, enorm, tmp, agg);
    k_gcn_combine<<<gNH, TB256, 0, stream>>>(agg, tmp, dis, b1, h);
    // --- GCN layer 2
    k_gemm_wmma<true><<<gemm128, 32, 0, stream>>>(h, W2, nullptr, tmp, HIDDEN);
    k_agg<<<gNode8, TB256, 0, stream>>>(offsets, csr, src, enorm, tmp, agg);
    k_gcn_combine<<<gNH, TB256, 0, stream>>>(agg, tmp, dis, b2, h);

    // --- cosine scores + per-node top-k keep mask
    k_nrm<<<gNode8, TB256, 0, stream>>>(h, nrm);
    k_scores<<<gEdge8, TB256, 0, stream>>>(src, dst, h, nrm, scores);
    k_topk<<<gN, TB256, 0, stream>>>(offsets, csr, scores, keepf);

    // --- GatedGraphConv (2 layers) on h0 with denoised edges
    for (int layer = 0; layer < 2; ++layer) {
        const float* Wl = Wg + (size_t)layer * HIDDEN * HIDDEN;
        float* xout = (layer == 1) ? (float*)d_out : x;
        k_gemm_wmma<false><<<gemm128, 32, 0, stream>>>(x, Wl, nullptr, tmp, HIDDEN);
        k_agg<<<gNode8, TB256, 0, stream>>>(offsets, csr, src, keepf, tmp, agg);
        k_gemm_wmma<true><<<gemm384, 32, 0, stream>>>(agg, Wih, bih, gi, H3);
        k_gemm_wmma<true><<<gemm384, 32, 0, stream>>>(x, Whh, bhh, gh, H3);
        k_gru<<<gNH, TB256, 0, stream>>>(gi, gh, x, xout);
    }
}